// AttentionFusion_15350213116457
// MI455X (gfx1250) — compile-verified
//
#include <hip/hip_runtime.h>

// ---------------------------------------------------------------------------
// AttentionFusion: alpha1 = sigmoid(x1.w - x2.w); fused = a1*x1 + (1-a1)*x2
// MI455X / gfx1250. Pure streaming op: ~810MB traffic, ~0.4 GFLOP
// (~0.5 flop/byte) -> HBM-bound, floor ~= 810MB / 23.3TB/s ~= 35us.
//
// WMMA intentionally unused: v_wmma contracts only along K, never along the
// lane axis, so the 128-wide dot products would need LDS/lane transposes that
// add traffic while saving zero HBM bytes (the only bottleneck). The CDNA5
// features this op rewards are used instead:
//   HBM --s_clause'd global_load_async_to_lds_b128 (ASYNCcnt, TH_LOAD_NT)--> LDS
//       --ds_load_b128--> VGPR --blend--> s_clause'd global_store_b128 TH_STORE_NT
// Triple-buffered per-wave staging (16 copies / 8KB in flight per wave),
// wave32 butterfly reduction for the dot products.
// ---------------------------------------------------------------------------

typedef float v4f __attribute__((ext_vector_type(4)));

#define D           128
#define ROW_BYTES   512                    // 128 f32
#define WAVES_PB    8
#define THREADS     (WAVES_PB * 32)
#define RPG         4                      // rows per group
#define GRP_BYTES   (RPG * ROW_BYTES)      // 2048 per matrix per group
#define NBUF        3                      // triple buffer
#define BUF_FLOATS  (RPG * 2 * D)          // 1024 floats = 4KB (x1+x2 halves)
#define WAVE_FLOATS (NBUF * BUF_FLOATS)    // 3072 floats = 12KB
#define SMEM_FLOATS (WAVES_PB * WAVE_FLOATS) // 24576 floats = 96KB

__global__ __launch_bounds__(THREADS)
void attn_fuse_kernel(const float* __restrict__ x1,
                      const float* __restrict__ x2,
                      const float* __restrict__ w,
                      float* __restrict__ fused,
                      float* __restrict__ alpha,
                      int n, int groupsPerWave) {
    __shared__ float smem[SMEM_FLOATS];

    const int lane  = threadIdx.x & 31;
    const int wave  = threadIdx.x >> 5;
    const long long gwave   = (long long)blockIdx.x * WAVES_PB + wave;
    const long long rowBase = gwave * (long long)groupsPerWave * RPG;

    float* wbuf = &smem[wave * WAVE_FLOATS];
    const unsigned ldsBaseU = (unsigned)(size_t)wbuf + (unsigned)lane * 16u;

    // weights: lane l owns columns 4l..4l+3; hot data -> default (RT) caching
    const v4f wv = reinterpret_cast<const v4f*>(w)[lane];

    // ---- software-pipelined issue state (running pointers, +2048B/group) ----
    long long          rowIss = rowBase;
    int                bufIss = 0;
    unsigned long long srcA = (unsigned long long)(size_t)x1 +
                              (unsigned long long)rowBase * ROW_BYTES + (unsigned)lane * 16u;
    unsigned long long srcB = (unsigned long long)(size_t)x2 +
                              (unsigned long long)rowBase * ROW_BYTES + (unsigned)lane * 16u;

    auto issueGroup = [&]() -> int {
        if (rowIss >= n) return 0;
        long long rem = (long long)n - rowIss;
        int rows = rem < RPG ? (int)rem : RPG;
        unsigned lds1 = ldsBaseU + (unsigned)(bufIss * (BUF_FLOATS * 4)); // x1 half
        unsigned lds2 = lds1 + (unsigned)(RPG * ROW_BYTES);               // x2 half
        if (rows == RPG) {
            // Single asm block: clause legality requires no interleaved
            // VALU/SALU. INST_OFFSET is added to BOTH LDS and global address.
            asm volatile(
                "s_clause 0x7\n\t"
                "global_load_async_to_lds_b128 %0, %1, off th:TH_LOAD_NT\n\t"
                "global_load_async_to_lds_b128 %0, %1, off offset:512 th:TH_LOAD_NT\n\t"
                "global_load_async_to_lds_b128 %0, %1, off offset:1024 th:TH_LOAD_NT\n\t"
                "global_load_async_to_lds_b128 %0, %1, off offset:1536 th:TH_LOAD_NT\n\t"
                "global_load_async_to_lds_b128 %2, %3, off th:TH_LOAD_NT\n\t"
                "global_load_async_to_lds_b128 %2, %3, off offset:512 th:TH_LOAD_NT\n\t"
                "global_load_async_to_lds_b128 %2, %3, off offset:1024 th:TH_LOAD_NT\n\t"
                "global_load_async_to_lds_b128 %2, %3, off offset:1536 th:TH_LOAD_NT"
                :: "v"(lds1), "v"(srcA), "v"(lds2), "v"(srcB) : "memory");
        } else {
            for (int r = 0; r < rows; ++r) {
                unsigned l1 = lds1 + (unsigned)(r * ROW_BYTES);
                unsigned l2 = lds2 + (unsigned)(r * ROW_BYTES);
                unsigned long long a1 = srcA + (unsigned long long)(r * ROW_BYTES);
                unsigned long long a2 = srcB + (unsigned long long)(r * ROW_BYTES);
                asm volatile("global_load_async_to_lds_b128 %0, %1, off th:TH_LOAD_NT" :: "v"(l1), "v"(a1) : "memory");
                asm volatile("global_load_async_to_lds_b128 %0, %1, off th:TH_LOAD_NT" :: "v"(l2), "v"(a2) : "memory");
            }
        }
        rowIss += RPG;
        srcA   += GRP_BYTES;
        srcB   += GRP_BYTES;
        bufIss  = (bufIss == NBUF - 1) ? 0 : bufIss + 1;
        return rows;
    };

    // prologue: stage current group + 1 lookahead group
    int cur    = issueGroup();   // group g
    int aheadA = issueGroup();   // group g+1

    unsigned long long dstF = (unsigned long long)(size_t)fused +
                              (unsigned long long)rowBase * ROW_BYTES + (unsigned)lane * 16u;
    float* alphaPtr = alpha + rowBase * 2;
    int    bufC     = 0;

    for (int g = 0; g < groupsPerWave; ++g) {
        if (cur == 0) break;
        const int aheadB = issueGroup();   // group g+2

        // async loads complete in order: with 16 (resp. 8) newer copies
        // outstanding, asynccnt<=16 (resp. <=8) proves group g is resident.
        if (aheadB == RPG) {
            asm volatile("s_wait_asynccnt 16" ::: "memory");
        } else if (aheadA == RPG) {
            asm volatile("s_wait_asynccnt 8" ::: "memory");
        } else {
            asm volatile("s_wait_asynccnt 0" ::: "memory");
        }

        // ---- compute from LDS buffer bufC ----
        const float* b  = &wbuf[bufC * BUF_FLOATS];
        const v4f*   pa = reinterpret_cast<const v4f*>(b) + lane;   // x1 rows
        const v4f*   pb = pa + (RPG * D / 4);                        // x2 rows

        v4f   xa[RPG], xb[RPG];
        float d[RPG];
#pragma unroll
        for (int r = 0; r < RPG; ++r) {
            xa[r] = pa[r * (D / 4)];
            xb[r] = pb[r * (D / 4)];
            float s1p = xa[r].x * wv.x + xa[r].y * wv.y + xa[r].z * wv.z + xa[r].w * wv.w;
            float s2p = xb[r].x * wv.x + xb[r].y * wv.y + xb[r].z * wv.z + xb[r].w * wv.w;
            d[r] = s1p - s2p;
        }
        // wave32 butterfly: every lane ends with the full dot difference
#pragma unroll
        for (int m = 16; m >= 1; m >>= 1) {
#pragma unroll
            for (int r = 0; r < RPG; ++r)
                d[r] += __shfl_xor(d[r], m, 32);
        }

        float a1[RPG], a2[RPG];
#pragma unroll
        for (int r = 0; r < RPG; ++r) {
            a1[r] = 1.0f / (1.0f + __expf(-d[r]));   // stable sigmoid
            a2[r] = 1.0f - a1[r];
        }

        v4f f[RPG];
#pragma unroll
        for (int r = 0; r < RPG; ++r) {
            f[r].x = a1[r] * xa[r].x + a2[r] * xb[r].x;
            f[r].y = a1[r] * xa[r].y + a2[r] * xb[r].y;
            f[r].z = a1[r] * xa[r].z + a2[r] * xb[r].z;
            f[r].w = a1[r] * xa[r].w + a2[r] * xb[r].w;
        }

        if (cur == RPG) {
            asm volatile(
                "s_clause 0x3\n\t"
                "global_store_b128 %0, %1, off th:TH_STORE_NT\n\t"
                "global_store_b128 %0, %2, off offset:512 th:TH_STORE_NT\n\t"
                "global_store_b128 %0, %3, off offset:1024 th:TH_STORE_NT\n\t"
                "global_store_b128 %0, %4, off offset:1536 th:TH_STORE_NT"
                :: "v"(dstF), "v"(f[0]), "v"(f[1]), "v"(f[2]), "v"(f[3]) : "memory");
            if (lane == 0) {
                float4* ap = reinterpret_cast<float4*>(alphaPtr);
                ap[0] = make_float4(a1[0], a2[0], a1[1], a2[1]);
                ap[1] = make_float4(a1[2], a2[2], a1[3], a2[3]);
            }
        } else {
            for (int r = 0; r < cur; ++r) {
                unsigned long long da = dstF + (unsigned long long)(r * ROW_BYTES);
                asm volatile("global_store_b128 %0, %1, off th:TH_STORE_NT" :: "v"(da), "v"(f[r]) : "memory");
            }
            if (lane == 0) {
                for (int r = 0; r < cur; ++r)
                    *reinterpret_cast<float2*>(alphaPtr + r * 2) = make_float2(a1[r], a2[r]);
            }
        }

        // advance pipeline
        dstF     += GRP_BYTES;
        alphaPtr += RPG * 2;
        bufC      = (bufC == NBUF - 1) ? 0 : bufC + 1;
        cur       = aheadA;
        aheadA    = aheadB;
    }
}

extern "C" void kernel_launch(void* const* d_in, const int* in_sizes, int n_in,
                              void* d_out, int out_size, void* d_ws, size_t ws_size,
                              hipStream_t stream) {
    (void)n_in; (void)out_size; (void)d_ws; (void)ws_size;
    const float* x1 = (const float*)d_in[0];
    const float* x2 = (const float*)d_in[1];
    const float* w  = (const float*)d_in[2];
    float* out = (float*)d_out;

    const int n = in_sizes[0] / D;            // N rows
    float* fused = out;                        // [N,128]
    float* alpha = out + (size_t)n * D;        // [N,2]

    const int GPW = 8;                         // groups (of 4 rows) per wave
    long long groups = ((long long)n + RPG - 1) / RPG;
    long long waves  = (groups + GPW - 1) / GPW;
    int blocks = (int)((waves + WAVES_PB - 1) / WAVES_PB);
    if (blocks < 1) blocks = 1;

    attn_fuse_kernel<<<blocks, THREADS, 0, stream>>>(x1, x2, w, fused, alpha, n, GPW);
}